// LocallyDense_59657095741740
// MI455X (gfx1250) — compile-verified
//
#include <hip/hip_runtime.h>

// Problem constants (from reference)
#define N_GROUPS   360
#define GROUP_SIZE 128
#define OUT_DIM    256
#define N_VOXELS   65536
#define BATCH      256
#define BN_EPS     1e-3f

#define BM        64     // batch rows per workgroup (4 WMMA M-tiles per wave)
#define A_STRIDE  132    // LDS row stride in floats (132 mod 64 = 4 -> conflict-free frag reads)

typedef float v2f __attribute__((ext_vector_type(2)));
typedef float v8f __attribute__((ext_vector_type(8)));

// V_WMMA_F32_16X16X4_F32: D = A(16x4) * B(4x16) + C(16x16), wave32 striped.
//  A frag: lane holds A[lane&15][kb..kb+1], kb = 4*kc + 2*(lane>>4)
//  B frag: lane holds B[kb..kb+1][lane&15]
//  C/D:    VGPR r -> row r + 8*(lane>>4), col lane&15
__launch_bounds__(256)
__global__ void locally_dense_wmma(
    const float* __restrict__ x,      // [BATCH][N_VOXELS]
    const int*   __restrict__ idx,    // [N_GROUPS][GROUP_SIZE]
    const float* __restrict__ W,      // [N_GROUPS][GROUP_SIZE][OUT_DIM]
    const float* __restrict__ bias,   // [N_GROUPS][OUT_DIM]
    const float* __restrict__ gamma,  // [OUT_DIM]
    const float* __restrict__ beta,   // [OUT_DIM]
    const float* __restrict__ mean,   // [OUT_DIM]
    const float* __restrict__ var,    // [OUT_DIM]
    float* __restrict__ out)          // [BATCH][N_GROUPS][OUT_DIM]
{
    __shared__ int   idx_s[GROUP_SIZE];
    __shared__ float A_s[BM][A_STRIDE];

    const int g    = blockIdx.x;        // group
    const int bm0  = blockIdx.y * BM;   // batch tile base
    const int tid  = threadIdx.x;
    const int lane = tid & 31;
    const int wave = tid >> 5;          // 0..7

    // ---- stage gather indices for this group ----
    if (tid < GROUP_SIZE) idx_s[tid] = idx[g * GROUP_SIZE + tid];
    __syncthreads();

    // ---- gather A = x[bm0:bm0+64, idx[g,:]] into LDS ----
    // Each thread owns one k (index read once), walks 32 of the 64 batch rows.
    {
        const int kk = tid & (GROUP_SIZE - 1);          // 0..127
        const int r0 = tid >> 7;                        // 0 or 1
        const float* __restrict__ xcol =
            x + (size_t)bm0 * N_VOXELS + idx_s[kk];
        #pragma unroll
        for (int i = 0; i < BM / 2; ++i) {
            const int r = 2 * i + r0;
            A_s[r][kk] = xcol[(size_t)r * N_VOXELS];
        }
    }
    __syncthreads();

    // wave w owns M-tiles {0..3} x N-tiles {2w, 2w+1}
    const int halfsel = lane >> 4;      // 0 or 1
    const int lan     = lane & 15;
    const int col0    = wave * 32 + lan;
    const int col1    = col0 + 16;

    const float* __restrict__ Wg = W + (size_t)g * GROUP_SIZE * OUT_DIM;

    v8f acc[4][2] = {};

    #pragma unroll 2
    for (int kc = 0; kc < GROUP_SIZE / 4; ++kc) {
        const int kb = kc * 4 + 2 * halfsel;   // this lane's K base within the chunk

        // B fragments from global (L2-resident W); reused by 4 M-tiles each
        const float* wp = Wg + (size_t)kb * OUT_DIM;
        v2f b0 = { wp[col0], wp[OUT_DIM + col0] };
        v2f b1 = { wp[col1], wp[OUT_DIM + col1] };

        #pragma unroll
        for (int mt = 0; mt < 4; ++mt) {
            // A fragment from LDS (b64 per lane, conflict-free via stride 132)
            v2f a = *(const v2f*)&A_s[mt * 16 + lan][kb];
            acc[mt][0] = __builtin_amdgcn_wmma_f32_16x16x4_f32(
                false, a, false, b0, (short)0, acc[mt][0], false, false);
            acc[mt][1] = __builtin_amdgcn_wmma_f32_16x16x4_f32(
                false, a, false, b1, (short)0, acc[mt][1], false, false);
        }
    }

    // ---- epilogue: fold bias + batchnorm affine (per-lane column fixed) ----
    // out = (acc + b) * inv + (beta - mean*inv),  inv = gamma * rsqrt(var + eps)
    const float inv0 = gamma[col0] * rsqrtf(var[col0] + BN_EPS);
    const float inv1 = gamma[col1] * rsqrtf(var[col1] + BN_EPS);
    const float add0 = bias[g * OUT_DIM + col0] * inv0 + (beta[col0] - mean[col0] * inv0);
    const float add1 = bias[g * OUT_DIM + col1] * inv1 + (beta[col1] - mean[col1] * inv1);

    float* __restrict__ outg = out + (size_t)g * OUT_DIM;
    #pragma unroll
    for (int mt = 0; mt < 4; ++mt) {
        #pragma unroll
        for (int r = 0; r < 8; ++r) {
            const int m = bm0 + mt * 16 + r + 8 * halfsel;     // global batch row
            const size_t rowoff = (size_t)m * ((size_t)N_GROUPS * OUT_DIM);
            outg[rowoff + col0] = acc[mt][0][r] * inv0 + add0;
            outg[rowoff + col1] = acc[mt][1][r] * inv1 + add1;
        }
    }
}

extern "C" void kernel_launch(void* const* d_in, const int* in_sizes, int n_in,
                              void* d_out, int out_size, void* d_ws, size_t ws_size,
                              hipStream_t stream) {
    (void)in_sizes; (void)n_in; (void)out_size; (void)d_ws; (void)ws_size;
    const float* x     = (const float*)d_in[0];
    const int*   idx   = (const int*)  d_in[1];
    const float* W     = (const float*)d_in[2];
    const float* b     = (const float*)d_in[3];
    const float* gamma = (const float*)d_in[4];
    const float* beta  = (const float*)d_in[5];
    const float* mean  = (const float*)d_in[6];
    const float* var   = (const float*)d_in[7];
    float* out = (float*)d_out;

    dim3 grid(N_GROUPS, BATCH / BM);   // (360, 4)
    locally_dense_wmma<<<grid, 256, 0, stream>>>(x, idx, W, b, gamma, beta, mean, var, out);
}